// GraphMIL_83442624627200
// MI455X (gfx1250) — compile-verified
//
#include <hip/hip_runtime.h>
#include <hip/hip_bf16.h>

// GraphMIL forward for MI455X (gfx1250, wave32).
// All dense GEMMs run on the matrix cores via V_WMMA_F32_16X16X4_F32 (fp32
// in/out, preserves reference numerics through LayerNorm/softmax).
// Message passing is a bandwidth-bound atomic scatter (coalesced 256-wide rows).
// Workspace: 3*N*H + AH*N + small  floats  (~155 MB).

typedef __attribute__((ext_vector_type(2))) float v2f;
typedef __attribute__((ext_vector_type(8))) float v8f;

constexpr int NN  = 50000;
constexpr int EE  = 800000;
constexpr int FF  = 768;
constexpr int HH  = 256;
constexpr int AA  = 128;
constexpr int AHH = 4;
constexpr int NCC = 7;

// ---------------------------------------------------------------------------
// Core: one block computes a 16-row x HO-col fp32 output panel with WMMA.
// Block = HO threads (HO/32 waves); wave w owns column tiles [32w,32w+32).
// A is LDS-staged in 64-wide K chunks (stride 68: bank-conflict-free, 8B
// aligned so the a-fragment reads become ds_load_b64). B streams from L2.
// Result panel lands in lds_out (stride HO+4) for a fused coalesced epilogue.
// ---------------------------------------------------------------------------
template<int K, int HO>
__device__ __forceinline__ void wmma_gemm_16rows(const float* __restrict__ A,
                                                 const float* __restrict__ B,
                                                 int m0, float* lds_a, float* lds_out)
{
    constexpr int KT  = 64;
    constexpr int KTP = 68;
    constexpr int HOP = HO + 4;
    constexpr int NT  = HO;              // threads per block
    const int tid   = threadIdx.x;
    const int lane  = tid & 31;
    const int wave  = tid >> 5;
    const int n0    = wave * 32;
    const int mrow  = lane & 15;
    const int khalf = (lane >> 4) << 1;  // lanes 16-31 hold K=2,3
    v8f acc0 = {};
    v8f acc1 = {};
    for (int k0 = 0; k0 < K; k0 += KT) {
        for (int idx = tid; idx < 16 * KT; idx += NT) {
            const int r = idx >> 6, c = idx & 63;
            lds_a[r * KTP + c] = A[(m0 + r) * K + k0 + c];
        }
        __syncthreads();
#pragma unroll
        for (int kk = 0; kk < KT; kk += 4) {
            v2f a, b0, b1;
            a.x = lds_a[mrow * KTP + kk + khalf];
            a.y = lds_a[mrow * KTP + kk + khalf + 1];
            const float* __restrict__ Bp = B + (k0 + kk + khalf) * HO;
            b0.x = Bp[n0 + mrow];
            b0.y = Bp[HO + n0 + mrow];
            b1.x = Bp[n0 + 16 + mrow];
            b1.y = Bp[HO + n0 + 16 + mrow];
            acc0 = __builtin_amdgcn_wmma_f32_16x16x4_f32(false, a, false, b0,
                                                         (short)0, acc0, false, false);
            acc1 = __builtin_amdgcn_wmma_f32_16x16x4_f32(false, a, false, b1,
                                                         (short)0, acc1, false, false);
        }
        __syncthreads();
    }
    // C/D layout: VGPR r, lanes 0-15 -> M=r, N=lane; lanes 16-31 -> M=r+8.
    const int mbase = (lane >> 4) << 3;
#pragma unroll
    for (int r = 0; r < 8; ++r) {
        lds_out[(mbase + r) * HOP + n0 + mrow]      = acc0[r];
        lds_out[(mbase + r) * HOP + n0 + 16 + mrow] = acc1[r];
    }
    __syncthreads();
}

// ------------------------- input projection: h = x@W_in + b_in -------------
__global__ void __launch_bounds__(256)
proj_kernel(const float* __restrict__ X, const float* __restrict__ W,
            const float* __restrict__ bias, float* __restrict__ out)
{
    __shared__ float lds_a[16 * 68];
    __shared__ float lds_out[16 * (HH + 4)];
    const int m0 = blockIdx.x * 16;
    wmma_gemm_16rows<FF, HH>(X, W, m0, lds_a, lds_out);
    const int tid = threadIdx.x;
    const float bb = bias[tid];
#pragma unroll
    for (int m = 0; m < 16; ++m)
        out[(m0 + m) * HH + tid] = lds_out[m * (HH + 4) + tid] + bb;
}

// ------------- GIN MLP stage 1: t1 = relu(LN(m@W1 + b1, g, b)) -------------
__global__ void __launch_bounds__(256)
gin_mlp1_kernel(const float* __restrict__ Min, const float* __restrict__ W,
                const float* __restrict__ bias, const float* __restrict__ g,
                const float* __restrict__ b, float* __restrict__ out)
{
    __shared__ float lds_a[16 * 68];
    __shared__ float lds_out[16 * (HH + 4)];
    __shared__ float mean_s[16], rstd_s[16];
    constexpr int HOP = HH + 4;
    const int m0 = blockIdx.x * 16;
    wmma_gemm_16rows<HH, HH>(Min, W, m0, lds_a, lds_out);
    const int tid = threadIdx.x;
    const float bb = bias[tid];
#pragma unroll
    for (int m = 0; m < 16; ++m) lds_out[m * HOP + tid] += bb;
    __syncthreads();
    if (tid < 16) {
        const float* row = lds_out + tid * HOP;
        float s = 0.f, sq = 0.f;
        for (int c = 0; c < HH; ++c) { const float v = row[c]; s += v; sq += v * v; }
        const float mean = s * (1.0f / HH);
        const float var  = sq * (1.0f / HH) - mean * mean;
        mean_s[tid] = mean;
        rstd_s[tid] = rsqrtf(var + 1e-5f);
    }
    __syncthreads();
    const float gg = g[tid], bo = b[tid];
#pragma unroll
    for (int m = 0; m < 16; ++m) {
        const float v = (lds_out[m * HOP + tid] - mean_s[m]) * rstd_s[m] * gg + bo;
        out[(m0 + m) * HH + tid] = fmaxf(v, 0.0f);
    }
}

// ------- GIN MLP stage 2: h_new = relu(LN(t1@W2 + b2, g, b)) + h_old -------
__global__ void __launch_bounds__(256)
gin_mlp2_kernel(const float* __restrict__ T1, const float* __restrict__ W,
                const float* __restrict__ bias, const float* __restrict__ g,
                const float* __restrict__ b, const float* __restrict__ hres,
                float* __restrict__ out)
{
    __shared__ float lds_a[16 * 68];
    __shared__ float lds_out[16 * (HH + 4)];
    __shared__ float mean_s[16], rstd_s[16];
    constexpr int HOP = HH + 4;
    const int m0 = blockIdx.x * 16;
    wmma_gemm_16rows<HH, HH>(T1, W, m0, lds_a, lds_out);
    const int tid = threadIdx.x;
    const float bb = bias[tid];
#pragma unroll
    for (int m = 0; m < 16; ++m) lds_out[m * HOP + tid] += bb;
    __syncthreads();
    if (tid < 16) {
        const float* row = lds_out + tid * HOP;
        float s = 0.f, sq = 0.f;
        for (int c = 0; c < HH; ++c) { const float v = row[c]; s += v; sq += v * v; }
        const float mean = s * (1.0f / HH);
        const float var  = sq * (1.0f / HH) - mean * mean;
        mean_s[tid] = mean;
        rstd_s[tid] = rsqrtf(var + 1e-5f);
    }
    __syncthreads();
    const float gg = g[tid], bo = b[tid];
#pragma unroll
    for (int m = 0; m < 16; ++m) {
        float v = (lds_out[m * HOP + tid] - mean_s[m]) * rstd_s[m] * gg + bo;
        v = fmaxf(v, 0.0f) + hres[(m0 + m) * HH + tid];
        out[(m0 + m) * HH + tid] = v;
    }
}

// --- attention: logits[k][n] = tanh(h@W1_k + b1_k) . W2_k + b2_k (HO=128) ---
__global__ void __launch_bounds__(128)
att_kernel(const float* __restrict__ Hm, const float* __restrict__ W1,
           const float* __restrict__ B1, const float* __restrict__ W2,
           const float* __restrict__ B2, float* __restrict__ logits)
{
    __shared__ float lds_a[16 * 68];
    __shared__ float lds_out[16 * (AA + 4)];
    constexpr int HOP = AA + 4;
    const int k  = blockIdx.y;
    const int m0 = blockIdx.x * 16;
    wmma_gemm_16rows<HH, AA>(Hm, W1 + k * HH * AA, m0, lds_a, lds_out);
    const int tid = threadIdx.x;
    const float bb = B1[k * AA + tid];
#pragma unroll
    for (int m = 0; m < 16; ++m) {
        const int idx = m * HOP + tid;
        lds_out[idx] = tanhf(lds_out[idx] + bb);
    }
    __syncthreads();
    if (tid < 16) {
        const float* __restrict__ w2 = W2 + k * AA;
        float acc = B2[k];
        const float* row = lds_out + tid * HOP;
        for (int a = 0; a < AA; ++a) acc += row[a] * w2[a];
        logits[k * NN + m0 + tid] = acc;
    }
}

// ------------------------- elementwise / scatter ---------------------------
__global__ void __launch_bounds__(256)
minit_kernel(const float* __restrict__ h, const float* __restrict__ eps,
             int layer, float* __restrict__ m)
{
    const int idx = blockIdx.x * 256 + threadIdx.x;
    m[idx] = (1.0f + eps[layer]) * h[idx];
}

__global__ void __launch_bounds__(256)
scatter_kernel(const float* __restrict__ h, const int* __restrict__ src,
               const int* __restrict__ dst, float* __restrict__ m)
{
    const int tid = threadIdx.x;   // = feature column, H=256
    const int e0  = blockIdx.x * 8;
#pragma unroll
    for (int i = 0; i < 8; ++i) {
        const int e = e0 + i;
        const int s = src[e], t = dst[e];
        atomicAdd(&m[t * HH + tid], h[s * HH + tid]);
    }
}

__global__ void __launch_bounds__(256)
zero_kernel(float* __restrict__ p, int n)
{
    const int i = blockIdx.x * 256 + threadIdx.x;
    if (i < n) p[i] = 0.0f;
}

// ---------------------- softmax over nodes (per head) ----------------------
__global__ void __launch_bounds__(256)
softmax_stats_kernel(const float* __restrict__ logits, float* __restrict__ stats)
{
    const int k = blockIdx.x;
    const float* __restrict__ L = logits + (size_t)k * NN;
    __shared__ float red[256];
    const int tid = threadIdx.x;
    float mx = -3.4e38f;
    for (int n = tid; n < NN; n += 256) mx = fmaxf(mx, L[n]);
    red[tid] = mx; __syncthreads();
    for (int s = 128; s > 0; s >>= 1) {
        if (tid < s) red[tid] = fmaxf(red[tid], red[tid + s]);
        __syncthreads();
    }
    mx = red[0]; __syncthreads();
    float sm = 0.f;
    for (int n = tid; n < NN; n += 256) sm += expf(L[n] - mx);
    red[tid] = sm; __syncthreads();
    for (int s = 128; s > 0; s >>= 1) {
        if (tid < s) red[tid] += red[tid + s];
        __syncthreads();
    }
    if (tid == 0) { stats[k * 2] = mx; stats[k * 2 + 1] = red[0]; }
}

// a[k][n] = softmax(logits); attn out [N,AH]; z[k][:] += a * h[n][:] --------
__global__ void __launch_bounds__(256)
attn_z_kernel(const float* __restrict__ logits, const float* __restrict__ stats,
              const float* __restrict__ h, float* __restrict__ out_attn,
              float* __restrict__ z)
{
    const int k   = blockIdx.y;
    const int n0  = blockIdx.x * 256;
    const int cnt = min(256, NN - n0);
    __shared__ float a_s[256];
    const float mx  = stats[k * 2];
    const float inv = 1.0f / stats[k * 2 + 1];
    const int tid = threadIdx.x;
    if (tid < cnt) {
        const float a = expf(logits[(size_t)k * NN + n0 + tid] - mx) * inv;
        a_s[tid] = a;
        out_attn[(size_t)(n0 + tid) * AHH + k] = a;
    }
    __syncthreads();
    float acc = 0.f;
    for (int i = 0; i < cnt; ++i) acc += a_s[i] * h[(size_t)(n0 + i) * HH + tid];
    atomicAdd(&z[k * HH + tid], acc);
}

// ----------------------------- classifier head -----------------------------
__global__ void __launch_bounds__(256)
classifier_kernel(const float* __restrict__ z,
                  const float* __restrict__ Wc1, const float* __restrict__ bc1,
                  const float* __restrict__ g1,  const float* __restrict__ b1,
                  const float* __restrict__ Wc2, const float* __restrict__ bc2,
                  const float* __restrict__ g2,  const float* __restrict__ b2,
                  const float* __restrict__ Wc3, const float* __restrict__ bc3,
                  float* __restrict__ probs)
{
    __shared__ float zag[HH], c1[128], c2[64], c3[8], stat[2];
    const int tid = threadIdx.x;
    zag[tid] = 0.25f * (z[tid] + z[HH + tid] + z[2 * HH + tid] + z[3 * HH + tid]);
    __syncthreads();
    if (tid < 128) {
        float acc = bc1[tid];
        for (int d = 0; d < HH; ++d) acc += zag[d] * Wc1[d * 128 + tid];
        c1[tid] = acc;
    }
    __syncthreads();
    if (tid == 0) {
        float s = 0.f, sq = 0.f;
        for (int j = 0; j < 128; ++j) { s += c1[j]; sq += c1[j] * c1[j]; }
        const float mean = s / 128.f, var = sq / 128.f - mean * mean;
        stat[0] = mean; stat[1] = rsqrtf(var + 1e-5f);
    }
    __syncthreads();
    if (tid < 128) c1[tid] = fmaxf((c1[tid] - stat[0]) * stat[1] * g1[tid] + b1[tid], 0.f);
    __syncthreads();
    if (tid < 64) {
        float acc = bc2[tid];
        for (int i = 0; i < 128; ++i) acc += c1[i] * Wc2[i * 64 + tid];
        c2[tid] = acc;
    }
    __syncthreads();
    if (tid == 0) {
        float s = 0.f, sq = 0.f;
        for (int j = 0; j < 64; ++j) { s += c2[j]; sq += c2[j] * c2[j]; }
        const float mean = s / 64.f, var = sq / 64.f - mean * mean;
        stat[0] = mean; stat[1] = rsqrtf(var + 1e-5f);
    }
    __syncthreads();
    if (tid < 64) c2[tid] = fmaxf((c2[tid] - stat[0]) * stat[1] * g2[tid] + b2[tid], 0.f);
    __syncthreads();
    if (tid < NCC) {
        float acc = bc3[tid];
        for (int i = 0; i < 64; ++i) acc += c2[i] * Wc3[i * NCC + tid];
        c3[tid] = acc;
    }
    __syncthreads();
    if (tid == 0) {
        float mx = c3[0];
        for (int j = 1; j < NCC; ++j) mx = fmaxf(mx, c3[j]);
        float e[NCC]; float s = 0.f;
        for (int j = 0; j < NCC; ++j) { e[j] = expf(c3[j] - mx); s += e[j]; }
        for (int j = 0; j < NCC; ++j) probs[j] = e[j] / s;
    }
}

// ---------------------------------------------------------------------------
extern "C" void kernel_launch(void* const* d_in, const int* in_sizes, int n_in,
                              void* d_out, int out_size, void* d_ws, size_t ws_size,
                              hipStream_t stream)
{
    const float* x       = (const float*)d_in[0];
    const int*   edge    = (const int*)  d_in[1];
    const float* W_in    = (const float*)d_in[2];
    const float* b_in    = (const float*)d_in[3];
    const float* gin_W1  = (const float*)d_in[4];
    const float* gin_b1  = (const float*)d_in[5];
    const float* gin_lng = (const float*)d_in[6];
    const float* gin_lnb = (const float*)d_in[7];
    const float* gin_W2  = (const float*)d_in[8];
    const float* gin_b2  = (const float*)d_in[9];
    const float* epsp    = (const float*)d_in[10];
    const float* ln_g    = (const float*)d_in[11];
    const float* ln_b    = (const float*)d_in[12];
    const float* att_W1  = (const float*)d_in[13];
    const float* att_b1  = (const float*)d_in[14];
    const float* att_W2  = (const float*)d_in[15];
    const float* att_b2  = (const float*)d_in[16];
    const float* Wc1     = (const float*)d_in[17];
    const float* bc1     = (const float*)d_in[18];
    const float* lnc1_g  = (const float*)d_in[19];
    const float* lnc1_b  = (const float*)d_in[20];
    const float* Wc2     = (const float*)d_in[21];
    const float* bc2     = (const float*)d_in[22];
    const float* lnc2_g  = (const float*)d_in[23];
    const float* lnc2_b  = (const float*)d_in[24];
    const float* Wc3     = (const float*)d_in[25];
    const float* bc3     = (const float*)d_in[26];

    const int* src = edge;
    const int* dst = edge + EE;

    float* ws = (float*)d_ws;
    const size_t NH = (size_t)NN * HH;
    float* hbuf   = ws;               // current node features
    float* bufA   = ws + NH;          // m / h_new ping-pong
    float* bufB   = ws + 2 * NH;      // t1
    float* logits = ws + 3 * NH;      // [AH, N]
    float* stats  = logits + (size_t)AHH * NN;  // [AH][max,sum]
    float* zbuf   = stats + 2 * AHH;            // [AH, H]

    const int GB = NN / 16;           // 3125 row panels (N divisible by 16)
    dim3 b256(256), b128(128);

    // h = x @ W_in + b_in
    proj_kernel<<<GB, b256, 0, stream>>>(x, W_in, b_in, hbuf);

    for (int i = 0; i < 2; ++i) {
        // m = (1+eps)*h  then scatter-add of neighbor rows
        minit_kernel<<<(int)(NH / 256), b256, 0, stream>>>(hbuf, epsp, i, bufA);
        scatter_kernel<<<EE / 8, b256, 0, stream>>>(hbuf, src, dst, bufA);
        gin_mlp1_kernel<<<GB, b256, 0, stream>>>(bufA, gin_W1 + i * HH * HH,
                                                 gin_b1 + i * HH, gin_lng + i * HH,
                                                 gin_lnb + i * HH, bufB);
        gin_mlp2_kernel<<<GB, b256, 0, stream>>>(bufB, gin_W2 + i * HH * HH,
                                                 gin_b2 + i * HH, ln_g + i * HH,
                                                 ln_b + i * HH, hbuf, bufA);
        float* t = hbuf; hbuf = bufA; bufA = t;   // h <- h_new
    }

    att_kernel<<<dim3(GB, AHH), b128, 0, stream>>>(hbuf, att_W1, att_b1,
                                                   att_W2, att_b2, logits);
    softmax_stats_kernel<<<AHH, b256, 0, stream>>>(logits, stats);
    zero_kernel<<<(AHH * HH + 255) / 256, b256, 0, stream>>>(zbuf, AHH * HH);

    float* outp = (float*)d_out;      // [7 probs][N*AH attn]
    attn_z_kernel<<<dim3((NN + 255) / 256, AHH), b256, 0, stream>>>(
        logits, stats, hbuf, outp + NCC, zbuf);
    classifier_kernel<<<1, b256, 0, stream>>>(zbuf, Wc1, bc1, lnc1_g, lnc1_b,
                                              Wc2, bc2, lnc2_g, lnc2_b,
                                              Wc3, bc3, outp);
}